// LinearSynthesizerInspiredMultiHeadAttention_43877385896446
// MI455X (gfx1250) — compile-verified
//
#include <hip/hip_runtime.h>
#include <hip/hip_bf16.h>

// ---------------------------------------------------------------------------
// LinearSynthesizer attention, algebraically restructured:
//   qh = sigmoid(q@wq+bq)  kh = sigmoid(k@wk+bk)          (B*S,16)
//   T(b,h,:)  = sum_s kh(b,s,h) * v(b,s,:)                (khT @ v, split-K)
//   kv(b,h,d) = T(b,h,:)@wv[:,h*64+d] + (sum_s kh)*bv
//   state     = cumsum_h kv ; W2(b,h,:) = state(b,h,:)@wo[h*64:(h+1)*64,:]
//   out       = qh @ W2(b) + bo
// All GEMMs use V_WMMA_F32_16X16X4_F32 (exact fp32).
// ---------------------------------------------------------------------------

typedef float v2f __attribute__((ext_vector_type(2)));
typedef float v8f __attribute__((ext_vector_type(8)));

#define BATCH 4
#define SEQ   4096
#define DIM   1024
#define HEADS 16
#define ROWS  (BATCH * SEQ)          // 16384
#define KSPLIT 16                    // split-K chunks for khT @ v

__device__ __forceinline__ v8f wmma_f32(v2f a, v2f b, v8f c) {
  return __builtin_amdgcn_wmma_f32_16x16x4_f32(false, a, false, b, (short)0, c, false, false);
}

__device__ __forceinline__ float sigmoidf(float x) {
  return 1.0f / (1.0f + __expf(-x));
}

// ---------------------------------------------------------------------------
// zero scratch region used by atomics (T + shsum)
__global__ void k_zero(float* __restrict__ p, int n) {
  int i = blockIdx.x * blockDim.x + threadIdx.x;
  if (i < n) p[i] = 0.0f;
}

// ---------------------------------------------------------------------------
// out(r,h) = sigmoid( X(r,:) @ W(:,h) + bias(h) ),  X:(ROWS,DIM) W:(DIM,16)
// one wave = one 16-row tile, full N=16; W staged in LDS.
__global__ void k_proj_sigmoid(const float* __restrict__ X,
                               const float* __restrict__ W,
                               const float* __restrict__ bias,
                               float* __restrict__ out) {
  __shared__ float lw[DIM * HEADS];                 // 64 KB
  for (int i = threadIdx.x; i < DIM * HEADS; i += blockDim.x) lw[i] = W[i];
  __syncthreads();

  const int wave = threadIdx.x >> 5;
  const int lane = threadIdx.x & 31;
  const int tile = blockIdx.x * (blockDim.x >> 5) + wave;   // 16-row tile id
  const int m    = lane & 15;                               // M (row) / N (col)
  const int hi   = lane >> 4;
  const int koff = hi << 1;                                 // K pair offset

  const float* xrow = X + (size_t)(tile * 16 + m) * DIM;

  v8f c = {};
  for (int kb = 0; kb < DIM; kb += 4) {
    v2f a = *(const v2f*)(xrow + kb + koff);                // A: M=m, K=kb+koff..+1
    v2f b;
    b.x = lw[(kb + koff + 0) * HEADS + m];                  // B: K, N=m
    b.y = lw[(kb + koff + 1) * HEADS + m];
    c = wmma_f32(a, b, c);
  }

  const int   rbase = tile * 16 + (hi << 3);
  const float bi    = bias[m];
#pragma unroll
  for (int r = 0; r < 8; ++r) {
    out[(size_t)(rbase + r) * HEADS + m] = sigmoidf(c[r] + bi);
  }
}

// ---------------------------------------------------------------------------
// shsum(b,h) = sum_s kh(b,s,h)
__global__ void k_colsum(const float* __restrict__ kh, float* __restrict__ shsum) {
  const int b = blockIdx.x >> 4, h = blockIdx.x & 15;
  const float* p = kh + (size_t)b * SEQ * HEADS + h;
  float s = 0.0f;
  for (int i = threadIdx.x; i < SEQ; i += 256) s += p[(size_t)i * HEADS];
  __shared__ float red[256];
  red[threadIdx.x] = s;
  __syncthreads();
  for (int w = 128; w > 0; w >>= 1) {
    if (threadIdx.x < w) red[threadIdx.x] += red[threadIdx.x + w];
    __syncthreads();
  }
  if (threadIdx.x == 0) shsum[blockIdx.x] = red[0];
}

// ---------------------------------------------------------------------------
// T(b,h,n) += sum_{s in chunk} kh(b,s,h) * v(b,s,n)   (split-K, f32 atomics)
// grid = (64 n-tiles, KSPLIT, BATCH), block = 1 wave.
__global__ void k_ktv_partial(const float* __restrict__ kh,
                              const float* __restrict__ V,
                              float* __restrict__ T) {
  const int b     = blockIdx.z;
  const int nbase = blockIdx.x * 16;
  const int s0    = blockIdx.y * (SEQ / KSPLIT);            // 256 s-values
  const int lane  = threadIdx.x & 31;
  const int m     = lane & 15;
  const int hi    = lane >> 4;
  const int koff  = hi << 1;

  const float* khb = kh + (size_t)b * SEQ * HEADS;
  const float* Vb  = V + (size_t)b * SEQ * DIM;
  float*       Tb  = T + (size_t)b * HEADS * DIM;

  v8f c = {};
  for (int kb = 0; kb < SEQ / KSPLIT; kb += 4) {
    const int s = s0 + kb + koff;
    v2f a, bb;
    a.x  = khb[(size_t)(s + 0) * HEADS + m];                // A: M=h=m, K=s
    a.y  = khb[(size_t)(s + 1) * HEADS + m];
    bb.x = Vb[(size_t)(s + 0) * DIM + nbase + m];           // B: K=s, N
    bb.y = Vb[(size_t)(s + 1) * DIM + nbase + m];
    c = wmma_f32(a, bb, c);
  }

#pragma unroll
  for (int r = 0; r < 8; ++r) {
    unsafeAtomicAdd(&Tb[(size_t)(r + (hi << 3)) * DIM + nbase + m], c[r]);
  }
}

// ---------------------------------------------------------------------------
// kv -> cumsum over heads -> W2.  one block per batch, 1024 threads.
__global__ void k_state_w2(const float* __restrict__ T,
                           const float* __restrict__ wv,
                           const float* __restrict__ bv,
                           const float* __restrict__ shsum,
                           const float* __restrict__ wo,
                           float* __restrict__ W2) {
  const int b   = blockIdx.x;
  const int tid = threadIdx.x;
  const int h   = tid >> 6, d = tid & 63;

  const float* Trow = T + ((size_t)b * HEADS + h) * DIM;
  float acc = 0.0f;
  for (int c = 0; c < DIM; ++c) acc += Trow[c] * wv[(size_t)c * DIM + h * 64 + d];
  acc += shsum[b * HEADS + h] * bv[h * 64 + d];

  __shared__ float kvs[1024];
  __shared__ float st[1024];
  kvs[tid] = acc;
  __syncthreads();
  float s = 0.0f;
  for (int hh = 0; hh <= h; ++hh) s += kvs[hh * 64 + d];    // cumsum over head axis
  st[tid] = s;
  __syncthreads();

  // W2(b,h2,n) = state(h2,:) . wo[h2*64 .. h2*64+63, n],  n = tid
  for (int h2 = 0; h2 < HEADS; ++h2) {
    const float* srow = st + h2 * 64;
    float w = 0.0f;
#pragma unroll 4
    for (int j = 0; j < 64; ++j) w += srow[j] * wo[(size_t)(h2 * 64 + j) * DIM + tid];
    W2[((size_t)b * HEADS + h2) * DIM + tid] = w;
  }
}

// ---------------------------------------------------------------------------
// out = qh @ W2(b) + bo.  one wave = 16 rows x 64 cols (4 WMMA tiles, A reuse)
__global__ void k_out(const float* __restrict__ qh,
                      const float* __restrict__ W2,
                      const float* __restrict__ bo,
                      float* __restrict__ out) {
  const int wave = threadIdx.x >> 5;
  const int lane = threadIdx.x & 31;
  const int gw   = blockIdx.x * (blockDim.x >> 5) + wave;   // 0..16383
  const int rt   = gw >> 4;                                 // 16-row tile (0..1023)
  const int ng   = gw & 15;                                 // 64-col group
  const int b    = rt >> 8;                                 // 256 row tiles / batch
  const int m    = lane & 15;
  const int hi   = lane >> 4;
  const int koff = hi << 1;

  const float* arow = qh + (size_t)(rt * 16 + m) * HEADS;
  const float* Wb   = W2 + (size_t)b * HEADS * DIM;
  const int    n0   = ng * 64 + m;

  v8f c0 = {}, c1 = {}, c2 = {}, c3 = {};
#pragma unroll
  for (int ks = 0; ks < 4; ++ks) {
    const int kb = ks * 4;
    v2f a = *(const v2f*)(arow + kb + koff);
    v2f b0, b1, b2, b3;
    const float* r0 = Wb + (size_t)(kb + koff + 0) * DIM + n0;
    const float* r1 = Wb + (size_t)(kb + koff + 1) * DIM + n0;
    b0.x = r0[0];  b0.y = r1[0];
    b1.x = r0[16]; b1.y = r1[16];
    b2.x = r0[32]; b2.y = r1[32];
    b3.x = r0[48]; b3.y = r1[48];
    c0 = wmma_f32(a, b0, c0);
    c1 = wmma_f32(a, b1, c1);
    c2 = wmma_f32(a, b2, c2);
    c3 = wmma_f32(a, b3, c3);
  }

  const int rbase = rt * 16 + (hi << 3);
#pragma unroll
  for (int r = 0; r < 8; ++r) {
    float* orow = out + (size_t)(rbase + r) * DIM + ng * 64 + m;
    orow[0]  = c0[r] + bo[ng * 64 + m +  0];
    orow[16] = c1[r] + bo[ng * 64 + m + 16];
    orow[32] = c2[r] + bo[ng * 64 + m + 32];
    orow[48] = c3[r] + bo[ng * 64 + m + 48];
  }
}

// ---------------------------------------------------------------------------
extern "C" void kernel_launch(void* const* d_in, const int* in_sizes, int n_in,
                              void* d_out, int out_size, void* d_ws, size_t ws_size,
                              hipStream_t stream) {
  const float* q  = (const float*)d_in[0];
  const float* k  = (const float*)d_in[1];
  const float* v  = (const float*)d_in[2];
  const float* wq = (const float*)d_in[3];
  const float* bq = (const float*)d_in[4];
  const float* wk = (const float*)d_in[5];
  const float* bk = (const float*)d_in[6];
  const float* wv = (const float*)d_in[7];
  const float* bv = (const float*)d_in[8];
  const float* wo = (const float*)d_in[9];
  const float* bo = (const float*)d_in[10];
  float* out = (float*)d_out;

  // workspace layout (floats)
  float* ws    = (float*)d_ws;
  float* qh    = ws;                                   // ROWS*16   = 262144
  float* kh    = qh + (size_t)ROWS * HEADS;            // 262144
  float* T     = kh + (size_t)ROWS * HEADS;            // B*16*1024 = 65536
  float* shsum = T + (size_t)BATCH * HEADS * DIM;      // 64
  float* W2    = shsum + BATCH * HEADS;                // 65536

  // 1) zero atomic-accumulated scratch (T + shsum)
  {
    int n = BATCH * HEADS * DIM + BATCH * HEADS;
    k_zero<<<(n + 255) / 256, 256, 0, stream>>>(T, n);
  }
  // 2) projections + sigmoid (WMMA f32)
  k_proj_sigmoid<<<ROWS / (16 * 8), 256, 0, stream>>>(q, wq, bq, qh);
  k_proj_sigmoid<<<ROWS / (16 * 8), 256, 0, stream>>>(k, wk, bk, kh);
  // 3) column sums of kh
  k_colsum<<<BATCH * HEADS, 256, 0, stream>>>(kh, shsum);
  // 4) T = khT @ v (split-K WMMA + f32 atomics)
  k_ktv_partial<<<dim3(DIM / 16, KSPLIT, BATCH), 32, 0, stream>>>(kh, v, T);
  // 5) kv -> cumsum(state) -> W2
  k_state_w2<<<BATCH, 1024, 0, stream>>>(T, wv, bv, shsum, wo, W2);
  // 6) out = qh @ W2 + bo (WMMA f32)
  k_out<<<(ROWS / 16) * 16 / 8, 256, 0, stream>>>(qh, W2, bo, out);
}